// HGATLayer_28114855919664
// MI455X (gfx1250) — compile-verified
//
#include <hip/hip_runtime.h>

#define NUM_HEADS 4
#define HEAD_DIM 64
#define BB 8
#define NN 1024
#define DD 256
#define NEG_INF_F (-9000000000000000.0f)

typedef float v2f __attribute__((ext_vector_type(2)));
typedef float v8f __attribute__((ext_vector_type(8)));
typedef int   v2i __attribute__((ext_vector_type(2)));
typedef int   v4i __attribute__((ext_vector_type(4)));

#define AS1 __attribute__((address_space(1)))
#define AS3 __attribute__((address_space(3)))

// Async global->LDS copy path (ASYNCcnt / s_wait_asynccnt). Builtins confirmed
// present by the round-2 probe; signatures take AS1 int-vector src pointers.
#if defined(__has_builtin)
#  if __has_builtin(__builtin_amdgcn_global_load_async_to_lds_b128) && \
      __has_builtin(__builtin_amdgcn_global_load_async_to_lds_b64) &&  \
      __has_builtin(__builtin_amdgcn_s_wait_asynccnt)
#    define HAVE_ASYNC_LDS 1
#  endif
#endif

__device__ __forceinline__ float leaky(float x, float slope) {
    return x >= 0.0f ? x : slope * x;
}

// ---------------------------------------------------------------------------
// Kernel 1: proj = X (8192x256) @ Wp (256x256), f32 WMMA 16x16x4
// grid (512, 2), block 256 (8 waves). Wave w owns N-tile blockIdx.y*128+w*16.
// ---------------------------------------------------------------------------
#define XT_STRIDE 260  // 16-row LDS tile, padded: 260 % 64 == 4 -> no bank conflicts

__global__ __launch_bounds__(256) void proj_gemm(const float* __restrict__ X,
                                                 const float* __restrict__ Wp,
                                                 float* __restrict__ proj) {
    __shared__ float xt[16 * XT_STRIDE];
    const int m0  = blockIdx.x * 16;
    const int tid = threadIdx.x;

    // stage the 16x256 X tile into LDS
#if defined(HAVE_ASYNC_LDS)
    {
        const int row = tid >> 4;
        const int cb  = (tid & 15) * 16;
        for (int j = 0; j < 4; ++j) {
            const int col = cb + j * 4;
            __builtin_amdgcn_global_load_async_to_lds_b128(
                (AS1 v4i*)(X + (size_t)(m0 + row) * DD + col),
                (AS3 v4i*)(&xt[row * XT_STRIDE + col]), 0, 0);
        }
        __builtin_amdgcn_s_wait_asynccnt(0);
    }
#else
    for (int i = tid; i < 16 * 256; i += 256) {
        const int r = i >> 8, c = i & 255;
        xt[r * XT_STRIDE + c] = X[(m0 + r) * DD + c];
    }
#endif
    __syncthreads();

    const int wave = tid >> 5;
    const int lane = tid & 31;
    const int half = lane >> 4;   // K-half select per ISA A/B layout
    const int l    = lane & 15;
    const int n0   = blockIdx.y * 128 + wave * 16;

    v8f acc = {};
    for (int k = 0; k < DD; k += 4) {
        v2f a, b;
        // A 16x4: lanes 0-15 -> K=k,k+1 ; lanes 16-31 -> K=k+2,k+3 (M = l)
        a.x = xt[l * XT_STRIDE + k + 2 * half];
        a.y = xt[l * XT_STRIDE + k + 2 * half + 1];
        // B 4x16: lanes 0-15 -> rows k,k+1 ; lanes 16-31 -> rows k+2,k+3 (N = l)
        b.x = Wp[(k + 2 * half) * DD + n0 + l];
        b.y = Wp[(k + 2 * half + 1) * DD + n0 + l];
        acc = __builtin_amdgcn_wmma_f32_16x16x4_f32(false, a, false, b,
                                                    (short)0, acc, false, false);
    }
    // D 16x16: VGPR r -> M = r + 8*half, N = l
    for (int r = 0; r < 8; ++r) {
        proj[(m0 + r + half * 8) * DD + n0 + l] = acc[r];
    }
}

// ---------------------------------------------------------------------------
// Kernel 2: s_l[row,h] = proj[row, h*64:..] . Ws[0:64]
//           s_r[row,h] = proj[row, h*64:..] . Ws[64:128]
// block 256 = 8 waves, one row per wave; 8 lanes per head, shuffle-reduce.
// ---------------------------------------------------------------------------
__global__ __launch_bounds__(256) void score_lr(const float* __restrict__ proj,
                                                const float* __restrict__ Ws,
                                                float* __restrict__ s_l,
                                                float* __restrict__ s_r) {
    const int tid  = threadIdx.x;
    const int wave = tid >> 5, lane = tid & 31;
    const int row  = blockIdx.x * 8 + wave;
    const int head = lane >> 3;  // 0..3
    const int lg   = lane & 7;   // lane within head group

    float dl = 0.0f, dr = 0.0f;
    for (int j = 0; j < 8; ++j) {
        const int d = lg * 8 + j;  // 0..63
        const float p = proj[row * DD + head * HEAD_DIM + d];
        dl += p * Ws[d];
        dr += p * Ws[HEAD_DIM + d];
    }
    dl += __shfl_xor(dl, 1); dr += __shfl_xor(dr, 1);
    dl += __shfl_xor(dl, 2); dr += __shfl_xor(dr, 2);
    dl += __shfl_xor(dl, 4); dr += __shfl_xor(dr, 4);
    if (lg == 0) {
        s_l[row * NUM_HEADS + head] = dl;
        s_r[row * NUM_HEADS + head] = dr;
    }
}

// ---------------------------------------------------------------------------
// Kernel 3: per (b, 16-row i-tile): scores -> softmax (unnormalized exp in
// LDS) -> msg = att @ proj_head via WMMA (8 waves = 2 heads x 4 N-tiles),
// fused residual + bias + leaky(0.5) epilogue.
// ---------------------------------------------------------------------------
#define ATT_STRIDE 1028  // 1028 % 64 == 4 -> conflict-free A-fragment reads

__global__ __launch_bounds__(256) void attention(const float* __restrict__ X,
                                                 const float* __restrict__ adj,
                                                 const float* __restrict__ proj,
                                                 const float* __restrict__ Ws,
                                                 const float* __restrict__ bias,
                                                 const float* __restrict__ s_l,
                                                 const float* __restrict__ s_r,
                                                 float* __restrict__ out) {
    __shared__ float att[2][16][ATT_STRIDE];  // ~131.6 KB (two heads)
    __shared__ float sr_s[NN * 2];            // s_r slice for the head pair (8 KB)
    __shared__ float red[2][16][16];
    __shared__ float rmax[2][16];
    __shared__ float rinv[2][16];

    const int b   = blockIdx.x >> 6;
    const int i0  = (blockIdx.x & 63) << 4;
    const int tid = threadIdx.x;
    const float wa = Ws[2 * HEAD_DIM];

    const int srow = tid >> 4;  // 0..15: score row
    const int sseg = tid & 15;  // 0..15: interleaved column group
    const int wave = tid >> 5;
    const int lane = tid & 31;
    const int half = lane >> 4;
    const int l    = lane & 15;

    for (int hp = 0; hp < 2; ++hp) {
        // ---- stage s_r[:, 2*hp : 2*hp+2] into LDS: sr_s[m*2 + hh]
#if defined(HAVE_ASYNC_LDS)
        for (int r = tid; r < NN; r += 256) {
            __builtin_amdgcn_global_load_async_to_lds_b64(
                (AS1 v2i*)(s_r + (size_t)(b * NN + r) * NUM_HEADS + hp * 2),
                (AS3 v2i*)(&sr_s[r * 2]), 0, 0);
        }
        __builtin_amdgcn_s_wait_asynccnt(0);
#else
        for (int r = tid; r < NN; r += 256) {
            sr_s[r * 2 + 0] = s_r[(b * NN + r) * NUM_HEADS + hp * 2 + 0];
            sr_s[r * 2 + 1] = s_r[(b * NN + r) * NUM_HEADS + hp * 2 + 1];
        }
#endif
        __syncthreads();

        // ---- score pass: two heads, interleaved columns (coalesced + bank-clean)
        for (int hh = 0; hh < 2; ++hh) {
            const int h = hp * 2 + hh;
            const float sl = s_l[(b * NN + i0 + srow) * NUM_HEADS + h];
            float lmax = -3.4e38f;
            for (int mm = 0; mm < 64; ++mm) {
                const int m = mm * 16 + sseg;
                const float a = adj[(b * NN + i0 + srow) * NN + m];
                float sc = sl + sr_s[m * 2 + hh] + a * wa;
                sc = leaky(sc, 0.2f);
                if (a <= 1e-5f) sc += NEG_INF_F;
                att[hh][srow][m] = sc;
                lmax = fmaxf(lmax, sc);
            }
            red[hh][srow][sseg] = lmax;
        }
        __syncthreads();
        if (tid < 32) {
            const int hh = tid >> 4, r = tid & 15;
            float m = red[hh][r][0];
            for (int j = 1; j < 16; ++j) m = fmaxf(m, red[hh][r][j]);
            rmax[hh][r] = m;
        }
        __syncthreads();
        // ---- exp pass (keep unnormalized; fold 1/sum after the GEMM)
        for (int hh = 0; hh < 2; ++hh) {
            const float m = rmax[hh][srow];
            float s = 0.0f;
            for (int mm = 0; mm < 64; ++mm) {
                const int c = mm * 16 + sseg;
                const float e = __expf(att[hh][srow][c] - m);
                att[hh][srow][c] = e;
                s += e;
            }
            red[hh][srow][sseg] = s;
        }
        __syncthreads();
        if (tid < 32) {
            const int hh = tid >> 4, r = tid & 15;
            float s = 0.0f;
            for (int j = 0; j < 16; ++j) s += red[hh][r][j];
            rinv[hh][r] = 1.0f / s;
        }
        __syncthreads();

        // ---- WMMA pass: msg(16x64) = att(16x1024) @ proj_head(1024x64)
        // wave -> (head-in-pair, N-tile); no divergence, EXEC all ones.
        const int hh = wave >> 2;
        const int h  = hp * 2 + hh;
        const int n0 = (wave & 3) * 16;
        v8f acc = {};
        for (int k = 0; k < NN; k += 4) {
            v2f a, bf;
            a.x  = att[hh][l][k + 2 * half];
            a.y  = att[hh][l][k + 2 * half + 1];
            bf.x = proj[(b * NN + k + 2 * half) * DD + h * HEAD_DIM + n0 + l];
            bf.y = proj[(b * NN + k + 2 * half + 1) * DD + h * HEAD_DIM + n0 + l];
            acc = __builtin_amdgcn_wmma_f32_16x16x4_f32(false, a, false, bf,
                                                        (short)0, acc, false, false);
        }
        const int feat = h * HEAD_DIM + n0 + l;
        for (int r = 0; r < 8; ++r) {
            const int row = r + half * 8;
            const int i   = i0 + row;
            const float v = acc[r] * rinv[hh][row];
            const float o = X[(b * NN + i) * DD + feat] + v + bias[feat];
            out[(b * NN + i) * DD + feat] = leaky(o, 0.5f);
        }
        __syncthreads();  // att / sr_s reused next head-pair
    }
}

// ---------------------------------------------------------------------------
extern "C" void kernel_launch(void* const* d_in, const int* in_sizes, int n_in,
                              void* d_out, int out_size, void* d_ws, size_t ws_size,
                              hipStream_t stream) {
    (void)in_sizes; (void)n_in; (void)out_size; (void)ws_size;
    const float* x    = (const float*)d_in[0];
    const float* adj  = (const float*)d_in[1];
    const float* Wp   = (const float*)d_in[2];
    const float* Ws   = (const float*)d_in[3];
    const float* bias = (const float*)d_in[4];
    float* out = (float*)d_out;

    float* proj = (float*)d_ws;                          // 8192*256 f32 (8 MB)
    float* s_l  = proj + (size_t)BB * NN * DD;           // 8192*4
    float* s_r  = s_l + (size_t)BB * NN * NUM_HEADS;     // 8192*4

    proj_gemm<<<dim3(BB * NN / 16, DD / 128), 256, 0, stream>>>(x, Wp, proj);
    score_lr<<<dim3(BB * NN / 8), 256, 0, stream>>>(proj, Ws, s_l, s_r);
    attention<<<dim3(BB * (NN / 16)), 256, 0, stream>>>(x, adj, proj, Ws, bias,
                                                        s_l, s_r, out);
    // tuple output: (out, adj) -> echo adj after out
    (void)hipMemcpyAsync(out + (size_t)BB * NN * DD, adj,
                         (size_t)BB * NN * NN * sizeof(float),
                         hipMemcpyDeviceToDevice, stream);
}